// K_graph_Layer_27702539059311
// MI455X (gfx1250) — compile-verified
//
#include <hip/hip_runtime.h>
#include <cstdint>
#include <cstddef>

typedef float v2f __attribute__((ext_vector_type(2)));
typedef float v8f __attribute__((ext_vector_type(8)));

#define C_DIM 16
#define H_DIM 128
#define NPAT  455      // C(15,3) patterns of "other 3 chosen columns"
#define EPSF  1e-5f

// rank of a 3-bit pattern (16-bit mask, bit c clear) in the canonical
// (a < b < d, bits remapped to 0..14 by deleting position c) enumeration.
__device__ __forceinline__ int patrank(int rem, int c) {
  int v[3] = {0, 0, 0};
  int j = 0;
  #pragma unroll
  for (int b = 0; b < 16; ++b) {
    if ((rem >> b) & 1) {
      int bb = b - (b > c ? 1 : 0);
      if (j < 3) v[j] = bb;
      ++j;
    }
  }
  int a = v[0], bq = v[1], d = v[2];
  return a + (bq * (bq - 1)) / 2 + (d * (d - 1) * (d - 2)) / 6;
}

// ---------------------------------------------------------------------------
// Stage 1: z = relu(X W1^T + b1); LayerNorm; logit = z_n . W2 + b2;
// fi = softmax_c; feat = X * fi; top-4 mask bits.
// One wave per n: 16 (c) x 128 (o) tile via 8 WMMA accumulators.
// ---------------------------------------------------------------------------
__global__ __launch_bounds__(256)
void k_stage1(const float* __restrict__ X, const float* __restrict__ W1,
              const float* __restrict__ b1, const float* __restrict__ lng,
              const float* __restrict__ lnb, const float* __restrict__ W2,
              const float* __restrict__ b2, float* __restrict__ feat,
              int* __restrict__ maskbits, int N) {
  const int tid  = threadIdx.x;
  const int lane = tid & 31;
  const int wave = tid >> 5;
  const int n    = blockIdx.x * 8 + wave;
  if (n >= N) return;

  const int m  = lane & 15;   // A-tile row (c) / B-tile column (o within tile)
  const int hi = lane >> 4;   // K half: lanes 0-15 -> K=0,1 ; 16-31 -> K=2,3

  const float* xrow = X + (size_t)(n * 16 + m) * H_DIM + 2 * hi;
  __builtin_prefetch(xrow, 0, 0);   // gfx1250 global_prefetch_b8

  const v8f vz = {0.f, 0.f, 0.f, 0.f, 0.f, 0.f, 0.f, 0.f};
  v8f acc[8];
  #pragma unroll
  for (int t = 0; t < 8; ++t) acc[t] = vz;

  #pragma unroll 4
  for (int k = 0; k < H_DIM; k += 4) {
    v2f a = *(const v2f*)(xrow + k);
    #pragma unroll
    for (int t = 0; t < 8; ++t) {
      // B[k'][o] = W1[o][k'], o = t*16 + m
      v2f b = *(const v2f*)(W1 + (size_t)(t * 16 + m) * H_DIM + 2 * hi + k);
      acc[t] = __builtin_amdgcn_wmma_f32_16x16x4_f32(
          false, a, false, b, (short)0, acc[t], false, false);
    }
  }

  // bias + relu; accumulate row sums for LayerNorm (row = c = r + 8*hi)
  float bsum[8], bsq[8];
  #pragma unroll
  for (int r = 0; r < 8; ++r) { bsum[r] = 0.f; bsq[r] = 0.f; }
  #pragma unroll
  for (int t = 0; t < 8; ++t) {
    float bb = b1[t * 16 + m];
    #pragma unroll
    for (int r = 0; r < 8; ++r) {
      float v = fmaxf(acc[t][r] + bb, 0.f);
      acc[t][r] = v;
      bsum[r] += v;
      bsq[r]  += v * v;
    }
  }
  // reduce across the 16 lanes of this half (o coverage), get mu / rstd per c
  #pragma unroll
  for (int r = 0; r < 8; ++r) {
    float s = bsum[r], q = bsq[r];
    #pragma unroll
    for (int msk = 1; msk < 16; msk <<= 1) {
      s += __shfl_xor(s, msk, 32);
      q += __shfl_xor(q, msk, 32);
    }
    float mu  = s * (1.f / 128.f);
    float var = q * (1.f / 128.f) - mu * mu;
    bsum[r] = mu;
    bsq[r]  = rsqrtf(var + EPSF);
  }

  // logits: sum over o of z_norm * W2[o]
  float lg[8];
  #pragma unroll
  for (int r = 0; r < 8; ++r) lg[r] = 0.f;
  #pragma unroll
  for (int t = 0; t < 8; ++t) {
    int o = t * 16 + m;
    float g = lng[o], bt = lnb[o], w2 = W2[o];
    #pragma unroll
    for (int r = 0; r < 8; ++r) {
      float zn = (acc[t][r] - bsum[r]) * bsq[r] * g + bt;
      lg[r] += zn * w2;
    }
  }
  const float b2v = b2[0];
  #pragma unroll
  for (int r = 0; r < 8; ++r) {
    #pragma unroll
    for (int msk = 1; msk < 16; msk <<= 1) lg[r] += __shfl_xor(lg[r], msk, 32);
    lg[r] += b2v;
  }
  // assemble all 16 logits per lane (other half via xor-16 shuffle)
  float l16[16];
  #pragma unroll
  for (int r = 0; r < 8; ++r) {
    float og = __shfl_xor(lg[r], 16, 32);
    l16[r]     = hi ? og : lg[r];
    l16[r + 8] = hi ? lg[r] : og;
  }

  // softmax over c
  float mx = l16[0];
  #pragma unroll
  for (int c = 1; c < 16; ++c) mx = fmaxf(mx, l16[c]);
  float fi[16], s = 0.f;
  #pragma unroll
  for (int c = 0; c < 16; ++c) { fi[c] = __expf(l16[c] - mx); s += fi[c]; }
  float invs = 1.f / s;
  #pragma unroll
  for (int c = 0; c < 16; ++c) fi[c] *= invs;

  // top-4 (strict >, first index wins == jax.lax.top_k tie rule)
  unsigned chosen = 0;
  #pragma unroll
  for (int k = 0; k < 4; ++k) {
    float bv = -1e30f;
    int bi = 0;
    #pragma unroll
    for (int c = 0; c < 16; ++c) {
      if (!((chosen >> c) & 1u) && fi[c] > bv) { bv = fi[c]; bi = c; }
    }
    chosen |= 1u << bi;
  }

  // feat = X * fi (each lane writes its float4 slice of every c-row)
  #pragma unroll
  for (int c = 0; c < 16; ++c) {
    const float4 xv = *(const float4*)(X + (size_t)(n * 16 + c) * H_DIM + lane * 4);
    float f = fi[c];
    float4 o;
    o.x = xv.x * f; o.y = xv.y * f; o.z = xv.z * f; o.w = xv.w * f;
    *(float4*)(feat + (size_t)(n * 16 + c) * H_DIM + lane * 4) = o;
  }
  if (lane == 0) maskbits[n] = (int)chosen;
}

// ---------------------------------------------------------------------------
// Stage 2a: scatter masked rows into per-(column, pattern) groups.
// ---------------------------------------------------------------------------
__global__ __launch_bounds__(128)
void k_scatter(const float* __restrict__ feat, const int* __restrict__ maskbits,
               int* __restrict__ rowcnt, int* __restrict__ rowlist,
               int* __restrict__ cnt, float* __restrict__ gsum, int N) {
  const int n = blockIdx.x;
  const int h = threadIdx.x;
  const int mb = maskbits[n];
  for (int c = 0; c < 16; ++c) {
    if (!((mb >> c) & 1)) continue;
    const int idx = patrank(mb & ~(1 << c), c);
    if (h == 0) {
      atomicAdd(&cnt[c * NPAT + idx], 1);
      int slot = atomicAdd(&rowcnt[c], 1);
      rowlist[c * N + slot] = n;
    }
    atomicAdd(&gsum[(size_t)(c * NPAT + idx) * H_DIM + h],
              feat[(size_t)(n * 16 + c) * H_DIM + h]);
  }
}

// ---------------------------------------------------------------------------
// Stage 2b: for every pattern q, agg[q] = (sum over patterns p with p∩q≠∅ of
// group sums) / deg, where deg = matching member count (== A row degree).
// ---------------------------------------------------------------------------
__global__ __launch_bounds__(128)
void k_patagg(const int* __restrict__ cnt, const float* __restrict__ gsum,
              float* __restrict__ aggq) {
  const int c = blockIdx.x / NPAT;
  const int q = blockIdx.x % NPAT;
  const int h = threadIdx.x;

  // unrank q -> bit triple (qa < qb < qd) in remapped 15-bit space
  int qa = 0, qb = 1, qd = 2;
  {
    int p = 0;
    for (int d = 2; d < 15; ++d)
      for (int b = 1; b < d; ++b)
        for (int a = 0; a < b; ++a) {
          if (p == q) { qa = a; qb = b; qd = d; }
          ++p;
        }
  }
  const int maskq = (1 << qa) | (1 << qb) | (1 << qd);

  float acc = 0.f;
  int tot = 0;
  int p = 0;
  for (int d = 2; d < 15; ++d)
    for (int b = 1; b < d; ++b)
      for (int a = 0; a < b; ++a) {
        const int mp = (1 << a) | (1 << b) | (1 << d);
        if (mp & maskq) {
          tot += cnt[c * NPAT + p];
          acc += gsum[(size_t)(c * NPAT + p) * H_DIM + h];
        }
        ++p;
      }
  const float inv = 1.f / fmaxf((float)tot, 1.f);
  aggq[(size_t)(c * NPAT + q) * H_DIM + h] = acc * inv;
}

// ---------------------------------------------------------------------------
// Stage 3: per column c, per 16-row tile of the compacted masked-row list:
// hc = relu(agg Wl^T + bl + feat Wr^T); store pre-norm hc; accumulate stats.
// One wave per tile; dual f32 WMMA GEMMs from LDS-staged A tiles.
// ---------------------------------------------------------------------------
__global__ __launch_bounds__(32)
void k_sage(const float* __restrict__ aggq, const float* __restrict__ feat,
            const int* __restrict__ rowcnt, const int* __restrict__ rowlist,
            const int* __restrict__ maskbits,
            const float* __restrict__ Wl, const float* __restrict__ bl,
            const float* __restrict__ Wr,
            float* __restrict__ Hpre, float* __restrict__ stats, int N) {
  const int tiles = (N + 15) >> 4;
  const int c    = blockIdx.x / tiles;
  const int tile = blockIdx.x % tiles;
  const int cnt_c = rowcnt[c];
  const int base  = tile * 16;
  if (base >= cnt_c) return;

  __shared__ float lA[16 * 132];   // 132-float pitch: bank-conflict-free
  __shared__ float lF[16 * 132];
  const int tid = threadIdx.x;

  {
    const int  s     = tid & 15;
    const bool doA   = tid < 16;
    const int  gs    = base + s;
    const bool valid = gs < cnt_c;
    const int  ns    = valid ? rowlist[c * N + gs] : 0;
    const float* src;
    if (doA) {
      int q = 0;
      if (valid) q = patrank(maskbits[ns] & ~(1 << c), c);
      src = aggq + (size_t)(c * NPAT + q) * H_DIM;
    } else {
      src = feat + (size_t)(ns * 16 + c) * H_DIM;
    }
    float* dst = (doA ? lA : lF) + s * 132;
    for (int hh = 0; hh < H_DIM; hh += 4) {
      float4 v = valid ? *(const float4*)(src + hh) : make_float4(0.f, 0.f, 0.f, 0.f);
      dst[hh] = v.x; dst[hh + 1] = v.y; dst[hh + 2] = v.z; dst[hh + 3] = v.w;
    }
  }
  __syncthreads();

  const int m   = tid & 15;
  const int hiH = tid >> 4;
  const v8f vz = {0.f, 0.f, 0.f, 0.f, 0.f, 0.f, 0.f, 0.f};
  v8f acc[8];
  #pragma unroll
  for (int t = 0; t < 8; ++t) acc[t] = vz;

  const float* WlC = Wl + (size_t)c * H_DIM * H_DIM;
  const float* WrC = Wr + (size_t)c * H_DIM * H_DIM;
  for (int k = 0; k < H_DIM; k += 4) {
    const int kk = k + 2 * hiH;
    v2f aA = *(const v2f*)(&lA[m * 132 + kk]);
    v2f aF = *(const v2f*)(&lF[m * 132 + kk]);
    #pragma unroll
    for (int t = 0; t < 8; ++t) {
      v2f bL = *(const v2f*)(WlC + (size_t)(t * 16 + m) * H_DIM + kk);
      acc[t] = __builtin_amdgcn_wmma_f32_16x16x4_f32(
          false, aA, false, bL, (short)0, acc[t], false, false);
      v2f bR = *(const v2f*)(WrC + (size_t)(t * 16 + m) * H_DIM + kk);
      acc[t] = __builtin_amdgcn_wmma_f32_16x16x4_f32(
          false, aF, false, bR, (short)0, acc[t], false, false);
    }
  }

  float bsum = 0.f, bsq = 0.f;
  #pragma unroll
  for (int t = 0; t < 8; ++t) {
    float bb = bl[c * H_DIM + t * 16 + m];
    #pragma unroll
    for (int r = 0; r < 8; ++r) {
      float v = fmaxf(acc[t][r] + bb, 0.f);
      const int slot = r + 8 * hiH;
      if (base + slot < cnt_c) {
        bsum += v;
        bsq  += v * v;
        Hpre[((size_t)c * N + base + slot) * H_DIM + t * 16 + m] = v;
      }
    }
  }
  #pragma unroll
  for (int msk = 1; msk < 32; msk <<= 1) {
    bsum += __shfl_xor(bsum, msk, 32);
    bsq  += __shfl_xor(bsq, msk, 32);
  }
  if (tid == 0) {
    atomicAdd(&stats[c], bsum);
    atomicAdd(&stats[16 + c], bsq);
  }
}

// ---------------------------------------------------------------------------
// Stage 4: finalize masked mean / rstd per column.
// ---------------------------------------------------------------------------
__global__ void k_stats(const float* __restrict__ stats,
                        const int* __restrict__ rowcnt,
                        float* __restrict__ murstd) {
  const int c = threadIdx.x;
  if (c >= 16) return;
  const float nel = fmaxf((float)(rowcnt[c] * H_DIM), 1.f);
  const float mu  = stats[c] / nel;
  const float var = stats[16 + c] / nel - mu * mu;
  murstd[c]      = mu;
  murstd[16 + c] = rsqrtf(var + EPSF);
}

// ---------------------------------------------------------------------------
// Stage 5: normalize and scatter hc rows to out[n, rank(c in sorted top-4), :]
// ---------------------------------------------------------------------------
__global__ __launch_bounds__(128)
void k_gather(const float* __restrict__ Hpre, const int* __restrict__ rowcnt,
              const int* __restrict__ rowlist, const int* __restrict__ maskbits,
              const float* __restrict__ murstd, float* __restrict__ out, int N) {
  const int c    = blockIdx.x / N;
  const int slot = blockIdx.x % N;
  if (slot >= rowcnt[c]) return;
  const int n = rowlist[c * N + slot];
  const int k = __popc((unsigned)maskbits[n] & ((1u << c) - 1u));
  const float mu = murstd[c], rs = murstd[16 + c];
  out[((size_t)n * 4 + k) * H_DIM + threadIdx.x] =
      (Hpre[((size_t)c * N + slot) * H_DIM + threadIdx.x] - mu) * rs;
}

// ---------------------------------------------------------------------------
extern "C" void kernel_launch(void* const* d_in, const int* in_sizes, int n_in,
                              void* d_out, int out_size, void* d_ws, size_t ws_size,
                              hipStream_t stream) {
  (void)n_in; (void)out_size; (void)ws_size;
  const float* X   = (const float*)d_in[0];
  const float* W1  = (const float*)d_in[1];
  const float* b1  = (const float*)d_in[2];
  const float* lng = (const float*)d_in[3];
  const float* lnb = (const float*)d_in[4];
  const float* W2  = (const float*)d_in[5];
  const float* b2  = (const float*)d_in[6];
  const float* Wl  = (const float*)d_in[7];
  const float* bl  = (const float*)d_in[8];
  const float* Wr  = (const float*)d_in[9];
  float* out = (float*)d_out;
  const int N = in_sizes[0] / (C_DIM * H_DIM);

  char* ws = (char*)d_ws;
  size_t p = 0;
  auto alloc = [&](size_t bytes) -> size_t {
    size_t r = p;
    p += (bytes + 255) & ~(size_t)255;
    return r;
  };
  float* feat    = (float*)(ws + alloc((size_t)N * C_DIM * H_DIM * 4));
  int*   maskb   = (int*)  (ws + alloc((size_t)N * 4));
  int*   rowlist = (int*)  (ws + alloc((size_t)16 * N * 4));
  float* aggq    = (float*)(ws + alloc((size_t)16 * NPAT * H_DIM * 4));
  float* Hpre    = (float*)(ws + alloc((size_t)16 * N * H_DIM * 4));
  float* murstd  = (float*)(ws + alloc(32 * 4));
  const size_t zstart = p;
  int*   rowcnt  = (int*)  (ws + alloc(16 * 4));
  int*   cnt     = (int*)  (ws + alloc((size_t)16 * NPAT * 4));
  float* gsum    = (float*)(ws + alloc((size_t)16 * NPAT * H_DIM * 4));
  float* stats   = (float*)(ws + alloc(32 * 4));
  const size_t zend = p;

  hipMemsetAsync(ws + zstart, 0, zend - zstart, stream);

  const int tiles = (N + 15) >> 4;
  k_stage1 <<<(N + 7) / 8,   256, 0, stream>>>(X, W1, b1, lng, lnb, W2, b2, feat, maskb, N);
  k_scatter<<<N,             128, 0, stream>>>(feat, maskb, rowcnt, rowlist, cnt, gsum, N);
  k_patagg <<<16 * NPAT,     128, 0, stream>>>(cnt, gsum, aggq);
  k_sage   <<<16 * tiles,     32, 0, stream>>>(aggq, feat, rowcnt, rowlist, maskb,
                                               Wl, bl, Wr, Hpre, stats, N);
  k_stats  <<<1,              16, 0, stream>>>(stats, rowcnt, murstd);
  k_gather <<<16 * N,        128, 0, stream>>>(Hpre, rowcnt, rowlist, maskb, murstd, out, N);
}